// MaskGeneratorNet_24197845745933
// MI455X (gfx1250) — compile-verified
//
#include <hip/hip_runtime.h>
#include <stdint.h>

typedef __attribute__((ext_vector_type(2))) float v2f;
typedef __attribute__((ext_vector_type(8))) float v8f;
typedef int v4i_t __attribute__((vector_size(16)));   // matches async-LDS builtin param

#define TOTALC    488418
#define OUT_TOTAL 4884180
#define KDIM      256

#if defined(__has_builtin)
#if __has_builtin(__builtin_amdgcn_global_load_async_to_lds_b128)
#define HAVE_ASYNC_LDS 1
#endif
#if __has_builtin(__builtin_amdgcn_s_wait_asynccnt)
#define WAIT_ASYNC(n) __builtin_amdgcn_s_wait_asynccnt(n)
#endif
#endif
#ifndef WAIT_ASYNC
#define WAIT_ASYNC(n) asm volatile("s_wait_asynccnt %0" ::"i"(n) : "memory")
#endif

__constant__ int SL_OFF[9]  = {0,160000,160400,320400,320800,480800,481200,488400,488418};
__constant__ int SL_N[8]    = {160000,400,160000,400,160000,400,7200,18};
__constant__ int SL_OB10[9] = {0,1600000,1604000,3204000,3208000,4808000,4812000,4884000,4884180};

// --- helpers -------------------------------------------------------------

// monotone map: larger float <-> larger uint
__device__ inline uint32_t fmono(float f) {
    uint32_t u = __float_as_uint(f);
    return (u & 0x80000000u) ? ~u : (u | 0x80000000u);
}

// hash-based gumbel noise, deterministic per element index
__device__ inline float gumbel_noise(uint32_t idx) {
    uint32_t s = idx * 747796405u + 2891336453u;
    uint32_t w = ((s >> ((s >> 28) + 4u)) ^ s) * 277803737u;
    w = (w >> 22) ^ w;
    float u = ((float)w + 0.5f) * (1.0f / 4294967296.0f); // (0,1)
    return -__logf(-__logf(u));
}

// --- kernel 1: h = relu(emb @ W1 + b1), stored transposed At[k][m], m padded to 16
__global__ __launch_bounds__(256) void k_hidden(const float* __restrict__ emb,
                                                const float* __restrict__ W1,
                                                const float* __restrict__ b1,
                                                float* __restrict__ At) {
    int t = blockIdx.x * 256 + threadIdx.x;   // 16 blocks -> 4096 threads
    if (t >= KDIM * 16) return;
    int k = t >> 4;
    int m = t & 15;
    float acc = 0.0f;
    if (m < 10) {
        acc = b1[k];
        #pragma unroll
        for (int i = 0; i < 10; ++i)
            acc += emb[m * 10 + i] * W1[i * KDIM + k];
        acc = fmaxf(acc, 0.0f);
    }
    At[k * 16 + m] = acc;
}

// --- kernel 2: z = h @ W2 + b2 + gumbel, scattered into concatenated-mask layout
__global__ __launch_bounds__(256) void k_gemm(const float* __restrict__ W2,
                                              const float* __restrict__ b2,
                                              const float* __restrict__ At,
                                              float* __restrict__ out) {
    __shared__ float sA[KDIM * 16];      // [k][m], 16 KB
    __shared__ float sB[3][32 * 128];    // triple-buffered B chunks, 48 KB

    const int tid   = threadIdx.x;
    const int lane  = tid & 31;
    const int wv    = tid >> 5;          // wave 0..7 -> 16-column subtile
    const int idx16 = lane & 15;         // m for A frag, n for B/C frags
    const int hi    = lane >> 4;         // K-pair selector within WMMA frag
    const int colbase = blockIdx.x * 128;
    const int col     = colbase + wv * 16 + idx16;
    const bool colok  = col < TOTALC;

    // stage A panel from workspace (4096 floats, b128 loads)
    {
        const float4* src = (const float4*)At;
        float4* dst = (float4*)sA;
        #pragma unroll
        for (int i = 0; i < 4; ++i)
            dst[tid + 256 * i] = src[tid + 256 * i];
    }

    const float bias = colok ? b2[col] : 0.0f;
    v8f acc = {};

    const bool fullTile = (colbase + 128) <= TOTALC;

    if (fullTile) {
#ifdef HAVE_ASYNC_LDS
        // --- async triple-buffered pipeline: global -> LDS direct, ASYNCcnt-tracked
        auto issue = [&](int kc, int b) {
            #pragma unroll
            for (int r = 0; r < 4; ++r) {
                const int row = wv * 4 + r;
                const float* gp = &W2[(size_t)(kc + row) * TOTALC + colbase + lane * 4];
                float* lp = &sB[b][row * 128 + lane * 4];
                __builtin_amdgcn_global_load_async_to_lds_b128(
                    (v4i_t*)gp, (v4i_t*)lp, /*offset=*/0, /*cpol=*/0);
            }
        };
        issue(0, 0);
        #pragma unroll
        for (int i = 0; i < 8; ++i) {
            if (i + 1 < 8) {
                issue((i + 1) * 32, (i + 1) % 3);   // 4 more in flight
                WAIT_ASYNC(4);                      // chunk i landed (in-order done)
            } else {
                WAIT_ASYNC(0);
            }
            __syncthreads();                        // all waves' chunk i visible
            const float* bp = &sB[i % 3][0];
            const int kc = i * 32;
            #pragma unroll
            for (int ks = 0; ks < 32; ks += 4) {
                const int kA = kc + ks + hi * 2;
                const int kB = ks + hi * 2;
                v2f a, b;
                a.x = sA[(kA + 0) * 16 + idx16];
                a.y = sA[(kA + 1) * 16 + idx16];
                b.x = bp[(kB + 0) * 128 + wv * 16 + idx16];
                b.y = bp[(kB + 1) * 128 + wv * 16 + idx16];
                acc = __builtin_amdgcn_wmma_f32_16x16x4_f32(
                    false, a, false, b, (short)0, acc, false, false);
            }
        }
#else
        // --- synchronous fallback (single buffer)
        for (int kc = 0; kc < KDIM; kc += 32) {
            __syncthreads();
            #pragma unroll
            for (int r = 0; r < 4; ++r) {
                int row = wv * 4 + r;
                const float* gp = &W2[(size_t)(kc + row) * TOTALC + colbase + lane * 4];
                *(float4*)(&sB[0][row * 128 + lane * 4]) = *(const float4*)gp;
                if (kc + 32 < KDIM)
                    __builtin_prefetch(gp + (size_t)32 * TOTALC, 0, 1);
            }
            __syncthreads();
            #pragma unroll
            for (int ks = 0; ks < 32; ks += 4) {
                const int kA = kc + ks + hi * 2;
                const int kB = ks + hi * 2;
                v2f a, b;
                a.x = sA[(kA + 0) * 16 + idx16];
                a.y = sA[(kA + 1) * 16 + idx16];
                b.x = sB[0][(kB + 0) * 128 + wv * 16 + idx16];
                b.y = sB[0][(kB + 1) * 128 + wv * 16 + idx16];
                acc = __builtin_amdgcn_wmma_f32_16x16x4_f32(
                    false, a, false, b, (short)0, acc, false, false);
            }
        }
#endif
    } else {
        // --- remainder tile (one block): guarded scalar staging
        for (int kc = 0; kc < KDIM; kc += 32) {
            __syncthreads();
            #pragma unroll
            for (int r = 0; r < 4; ++r) {
                int row = wv * 4 + r;
                #pragma unroll
                for (int e = 0; e < 4; ++e) {
                    int c2 = colbase + lane * 4 + e;
                    sB[0][row * 128 + lane * 4 + e] =
                        (c2 < TOTALC) ? W2[(size_t)(kc + row) * TOTALC + c2] : 0.0f;
                }
            }
            __syncthreads();
            #pragma unroll
            for (int ks = 0; ks < 32; ks += 4) {
                const int kA = kc + ks + hi * 2;
                const int kB = ks + hi * 2;
                v2f a, b;
                a.x = sA[(kA + 0) * 16 + idx16];
                a.y = sA[(kA + 1) * 16 + idx16];
                b.x = sB[0][(kB + 0) * 128 + wv * 16 + idx16];
                b.y = sB[0][(kB + 1) * 128 + wv * 16 + idx16];
                acc = __builtin_amdgcn_wmma_f32_16x16x4_f32(
                    false, a, false, b, (short)0, acc, false, false);
            }
        }
    }

    if (!colok) return;

    // slice lookup once per lane (col fixed)
    int s = 0;
    #pragma unroll
    for (int i = 1; i < 8; ++i)
        if (col >= SL_OFF[i]) s = i;
    const int ns    = SL_N[s];
    const int jcol  = col - SL_OFF[s];
    const int obase = SL_OB10[s];

    // C/D layout: lanes 0-15 hold M=v, lanes 16-31 hold M=v+8, N = lane&15
    #pragma unroll
    for (int v = 0; v < 8; ++v) {
        int m = v + hi * 8;
        if (m < 10) {
            float z = acc[v] + bias + gumbel_noise((uint32_t)(m * TOTALC + col));
            out[obase + m * ns + jcol] = z;
        }
    }
}

// --- kernel 3: per (batch,slice) radix-select the k-th largest (k = n/2)
__global__ __launch_bounds__(256) void k_select(const float* __restrict__ zout,
                                                uint32_t* __restrict__ thresh) {
    const int g = blockIdx.x;       // 0..79
    const int b = g >> 3;
    const int s = g & 7;
    const int n = SL_N[s];
    const int base = SL_OB10[s] + b * n;

    __shared__ uint32_t hist[256];
    __shared__ uint32_t sh_prefix, sh_k;
    if (threadIdx.x == 0) { sh_prefix = 0u; sh_k = (uint32_t)(n >> 1); }

    for (int p = 0; p < 4; ++p) {
        const int bp = 3 - p;
        hist[threadIdx.x] = 0u;
        __syncthreads();
        const uint32_t prefix = sh_prefix;
        for (int j = threadIdx.x; j < n; j += 256) {
            uint32_t u = fmono(zout[base + j]);
            uint32_t high = (bp == 3) ? 0u : (u >> ((bp + 1) * 8));
            if (high == prefix)
                atomicAdd(&hist[(u >> (bp * 8)) & 0xFFu], 1u);
        }
        __syncthreads();
        if (threadIdx.x == 0) {
            uint32_t kk = sh_k, cum = 0;
            int bin = 255;
            for (; bin > 0; --bin) {
                cum += hist[bin];
                if (cum >= kk) break;
            }
            if (cum < kk) cum += hist[0];           // bin == 0 fall-through
            sh_k = kk - (cum - hist[bin]);
            sh_prefix = (prefix << 8) | (uint32_t)bin;
        }
        __syncthreads();
    }
    if (threadIdx.x == 0) thresh[g] = sh_prefix;   // exact k-th largest key
}

// --- kernel 4: in-place hard mask: out = (key >= thresh) ? 1 : 0
__global__ __launch_bounds__(256) void k_mask(float* __restrict__ out,
                                              const uint32_t* __restrict__ thresh) {
    int idx = blockIdx.x * 256 + threadIdx.x;
    if (idx >= OUT_TOTAL) return;
    int s = 0;
    #pragma unroll
    for (int i = 1; i < 8; ++i)
        if (idx >= SL_OB10[i]) s = i;
    const int rem = idx - SL_OB10[s];
    const int b   = rem / SL_N[s];
    const uint32_t t = thresh[b * 8 + s];
    const uint32_t u = fmono(out[idx]);
    out[idx] = (u >= t) ? 1.0f : 0.0f;
}

extern "C" void kernel_launch(void* const* d_in, const int* in_sizes, int n_in,
                              void* d_out, int out_size, void* d_ws, size_t ws_size,
                              hipStream_t stream) {
    // setup_inputs order: x, embedding_input, W1, b1, W2, b2 (all fp32)
    const float* emb = (const float*)d_in[1];
    const float* W1  = (const float*)d_in[2];
    const float* b1  = (const float*)d_in[3];
    const float* W2  = (const float*)d_in[4];
    const float* b2  = (const float*)d_in[5];
    float* out = (float*)d_out;

    float*    At     = (float*)d_ws;                          // 16 KB: A^T panel
    uint32_t* thresh = (uint32_t*)((char*)d_ws + 16384);      // 320 B: thresholds

    hipLaunchKernelGGL(k_hidden, dim3(16), dim3(256), 0, stream, emb, W1, b1, At);
    hipLaunchKernelGGL(k_gemm, dim3((TOTALC + 127) / 128), dim3(256), 0, stream,
                       W2, b2, At, out);
    hipLaunchKernelGGL(k_select, dim3(80), dim3(256), 0, stream, out, thresh);
    hipLaunchKernelGGL(k_mask, dim3((OUT_TOTAL + 255) / 256), dim3(256), 0, stream,
                       out, thresh);
}